// GARCH11_28346784154009
// MI455X (gfx1250) — compile-verified
//
#include <hip/hip_runtime.h>

typedef __attribute__((ext_vector_type(2))) float v2f;
typedef __attribute__((ext_vector_type(8))) float v8f;

#define T_TOTAL 16777216              // 2^24
#define N_SCAN  (T_TOTAL - 1)         // length of the recurrence tail
#define SEG     4096                  // elements per segment (one wave each)
#define NSEG    4096                  // SEG*NSEG == T_TOTAL >= N_SCAN
#define TILE    256                   // elements per WMMA tile (16x16)
#define TILES_PER_SEG (SEG / TILE)    // 16
#define NB_RED  1024                  // blocks in variance reduction pass 1

// ws layout (float indices)
#define WS_S0    0
#define WS_CARRY 16
#define WS_START (16 + NSEG)
#define WS_PART  (16 + 2 * NSEG)

__device__ __forceinline__ float readlane_f(float v, int lane) {
    return __int_as_float(__builtin_amdgcn_readlane(__float_as_int(v), lane));
}

__device__ __forceinline__ float powB(float b, int p) {
    float r = 1.0f;
    for (int i = 0; i < p; ++i) r *= b;
    return r;
}

// ---------------------------------------------------------------------------
// One 256-element tile: Y = L * D via 4 chained fp32 WMMA 16x16x4 ops,
// L[i][j] = beta^(i-j) lower-triangular Toeplitz (held in aFrag).
// s (wave-uniform) is the value entering the tile; returns value after it.
// GUARD: bounds-check loads/stores against N_SCAN (only last tile of grid).
// ---------------------------------------------------------------------------
template <bool STORE, bool GUARD>
__device__ __forceinline__ float run_tile(
    const float* __restrict__ r, float* __restrict__ out, int g0, float s,
    float omega, float alpha, const v2f* aFrag, const float* betaP,
    float beta16, int n, int hi)
{
    v8f acc = {0.f, 0.f, 0.f, 0.f, 0.f, 0.f, 0.f, 0.f};
#pragma unroll
    for (int kk = 0; kk < 4; ++kk) {
        // B fragment: B[K=kk*4+2*hi + {0,1}][N=n] = d[g0 + n*16 + K]
        const int idx = g0 + n * 16 + 4 * kk + 2 * hi;
        float r0, r1;
        if (GUARD) {
            r0 = (idx     < N_SCAN) ? r[idx]     : 0.f;
            r1 = (idx + 1 < N_SCAN) ? r[idx + 1] : 0.f;
        } else {
            r0 = r[idx];
            r1 = r[idx + 1];
        }
        v2f b;
        b.x = omega + alpha * r0 * r0;
        b.y = omega + alpha * r1 * r1;
        if (GUARD) {   // padded elements contribute d = 0 (pure decay)
            b.x = (idx     < N_SCAN) ? b.x : 0.f;
            b.y = (idx + 1 < N_SCAN) ? b.y : 0.f;
        }
        acc = __builtin_amdgcn_wmma_f32_16x16x4_f32(
            /*neg_a=*/false, aFrag[kk], /*neg_b=*/false, b,
            /*c_mod=*/(short)0, acc, /*reuse_a=*/false, /*reuse_b=*/false);
    }

    // Stitch 16 sub-chunks: s_init[c+1] = Y[15][c] + beta^16 * s_init[c].
    // Y[15][c] lives in acc[7] of lane 16+c -> uniform SALU fmac chain.
    float sinit = 0.f;
    float cur = s;
#pragma unroll
    for (int c = 0; c < 16; ++c) {
        sinit = (n == c) ? cur : sinit;
        float e = readlane_f(acc[7], 16 + c);
        cur = __builtin_fmaf(beta16, cur, e);
    }

    if (STORE) {
        // Lane owns rows m = v + 8*hi of column n -> 8 consecutive elems.
        const int k0 = g0 + n * 16 + 8 * hi;
#pragma unroll
        for (int v = 0; v < 8; ++v) {
            const float val = __builtin_fmaf(betaP[v], sinit, acc[v]);
            const int k = k0 + v;
            if (!GUARD || k < N_SCAN) out[1 + k] = val;
        }
    }
    return cur;
}

// Scan one segment of SEG elements of d[k] = omega + alpha*r[k]^2 with the
// recurrence s = d + beta*s, starting value s (value *before* first elem).
// Only the final tile of the final segment needs bounds guards; every other
// tile runs a branch-free body.
template <bool STORE>
__device__ __forceinline__ float run_segment(
    const float* __restrict__ r, float* __restrict__ out, int seg, float s,
    float omega, float alpha, const v2f* aFrag, const float* betaP, float beta16)
{
    const int lane = threadIdx.x & 31;
    const int n  = lane & 15;
    const int hi = lane >> 4;
    const int segBase = seg * SEG;
    const bool last = (seg == NSEG - 1);
    const int nFull = last ? (TILES_PER_SEG - 1) : TILES_PER_SEG;

    int t = 0;
#pragma unroll 1
    for (; t < nFull; ++t)
        s = run_tile<STORE, false>(r, out, segBase + t * TILE, s,
                                   omega, alpha, aFrag, betaP, beta16, n, hi);
    if (last)
        s = run_tile<STORE, true>(r, out, segBase + t * TILE, s,
                                  omega, alpha, aFrag, betaP, beta16, n, hi);
    return s;
}

// Build per-lane A fragments (L slices) and beta powers once per wave.
__device__ __forceinline__ void build_frags(float beta, v2f* aFrag,
                                            float* betaP, float& beta16)
{
    const int lane = threadIdx.x & 31;
    const int m  = lane & 15;
    const int hi = lane >> 4;
#pragma unroll
    for (int kk = 0; kk < 4; ++kk) {
        const int j0 = 4 * kk + 2 * hi;
        v2f a;
        a.x = (m >= j0)     ? powB(beta, m - j0)     : 0.f;
        a.y = (m >= j0 + 1) ? powB(beta, m - j0 - 1) : 0.f;
        aFrag[kk] = a;
    }
    float p = powB(beta, 8 * hi + 1);
#pragma unroll
    for (int v = 0; v < 8; ++v) { betaP[v] = p; p *= beta; }
    beta16 = powB(beta, 16);
}

// ---------------- Pass 1a: partial sums for variance -----------------------
__global__ void __launch_bounds__(256)
k_var_partial(const float* __restrict__ r, float* __restrict__ ws)
{
    __shared__ float ssum[256];
    __shared__ float ssq[256];
    const int tid = threadIdx.x;
    const int stride = gridDim.x * 256;
    float s = 0.f, q = 0.f;
    for (int i = blockIdx.x * 256 + tid; i < T_TOTAL; i += stride) {
        const float v = r[i];
        s += v;
        q = __builtin_fmaf(v, v, q);
    }
    ssum[tid] = s; ssq[tid] = q;
    __syncthreads();
    for (int off = 128; off > 0; off >>= 1) {
        if (tid < off) { ssum[tid] += ssum[tid + off]; ssq[tid] += ssq[tid + off]; }
        __syncthreads();
    }
    if (tid == 0) {
        ws[WS_PART + 2 * blockIdx.x]     = ssum[0];
        ws[WS_PART + 2 * blockIdx.x + 1] = ssq[0];
    }
}

// ---------------- Pass 1b: finalize s0 = var(returns, ddof=1) --------------
__global__ void __launch_bounds__(256)
k_var_final(float* __restrict__ ws)
{
    __shared__ double dsum[256];
    __shared__ double dsq[256];
    const int tid = threadIdx.x;
    double s = 0.0, q = 0.0;
    for (int j = tid; j < NB_RED; j += 256) {
        s += (double)ws[WS_PART + 2 * j];
        q += (double)ws[WS_PART + 2 * j + 1];
    }
    dsum[tid] = s; dsq[tid] = q;
    __syncthreads();
    for (int off = 128; off > 0; off >>= 1) {
        if (tid < off) { dsum[tid] += dsum[tid + off]; dsq[tid] += dsq[tid + off]; }
        __syncthreads();
    }
    if (tid == 0) {
        const double mean = dsum[0] / (double)T_TOTAL;
        const double var  = (dsq[0] - dsum[0] * mean) / (double)(T_TOTAL - 1);
        ws[WS_S0] = (float)var;
    }
}

// ---------------- Pass 2: per-segment carries (zero-init scan end) ---------
__global__ void __launch_bounds__(32)
k_carry(const float* __restrict__ r, const float* __restrict__ pOmega,
        const float* __restrict__ pAlpha, const float* __restrict__ pBeta,
        float* __restrict__ ws)
{
    const float omega = pOmega[0], alpha = pAlpha[0], beta = pBeta[0];
    v2f aFrag[4]; float betaP[8]; float beta16;
    build_frags(beta, aFrag, betaP, beta16);
    const float c = run_segment<false>(r, nullptr, blockIdx.x, 0.f,
                                       omega, alpha, aFrag, betaP, beta16);
    if ((threadIdx.x & 31) == 0) ws[WS_CARRY + blockIdx.x] = c;
}

// ---------------- Pass 3: cross-segment scan (single wave) -----------------
__global__ void __launch_bounds__(32)
k_segscan(const float* __restrict__ pBeta, float* __restrict__ ws)
{
    const int lane = threadIdx.x & 31;
    const float beta = pBeta[0];
    float A = beta;                       // beta^SEG = beta^(2^12)
    for (int i = 0; i < 12; ++i) A = A * A;
    float Ap = A;                         // A^128 = A^(2^7)
    for (int i = 0; i < 7; ++i) Ap = Ap * Ap;

    const int PER = NSEG / 32;            // 128 segments per lane
    const int base = lane * PER;

    float Q = 0.f;                        // lane-local additive composition
    for (int j = 0; j < PER; ++j)
        Q = __builtin_fmaf(A, Q, ws[WS_CARRY + base + j]);

    const float s0 = ws[WS_S0];
    float start = 0.f;
    float cur = s0;
#pragma unroll
    for (int c = 0; c < 32; ++c) {
        start = (lane == c) ? cur : start;
        cur = __builtin_fmaf(Ap, cur, readlane_f(Q, c));
    }

    float S = start;
    for (int j = 0; j < PER; ++j) {
        ws[WS_START + base + j] = S;      // value entering segment base+j
        S = __builtin_fmaf(A, S, ws[WS_CARRY + base + j]);
    }
}

// ---------------- Pass 4: final scan + store -------------------------------
__global__ void __launch_bounds__(32)
k_final(const float* __restrict__ r, const float* __restrict__ pOmega,
        const float* __restrict__ pAlpha, const float* __restrict__ pBeta,
        const float* __restrict__ ws, float* __restrict__ out)
{
    const float omega = pOmega[0], alpha = pAlpha[0], beta = pBeta[0];
    v2f aFrag[4]; float betaP[8]; float beta16;
    build_frags(beta, aFrag, betaP, beta16);
    const float s = ws[WS_START + blockIdx.x];
    run_segment<true>(r, out, blockIdx.x, s, omega, alpha, aFrag, betaP, beta16);
    if (blockIdx.x == 0 && threadIdx.x == 0) out[0] = ws[WS_S0];   // s0
}

extern "C" void kernel_launch(void* const* d_in, const int* in_sizes, int n_in,
                              void* d_out, int out_size, void* d_ws, size_t ws_size,
                              hipStream_t stream)
{
    (void)in_sizes; (void)n_in; (void)out_size; (void)ws_size;
    const float* r      = (const float*)d_in[0];
    const float* pOmega = (const float*)d_in[1];
    const float* pAlpha = (const float*)d_in[2];
    const float* pBeta  = (const float*)d_in[3];
    float* out = (float*)d_out;
    float* ws  = (float*)d_ws;

    k_var_partial<<<NB_RED, 256, 0, stream>>>(r, ws);
    k_var_final  <<<1,      256, 0, stream>>>(ws);
    k_carry      <<<NSEG,    32, 0, stream>>>(r, pOmega, pAlpha, pBeta, ws);
    k_segscan    <<<1,       32, 0, stream>>>(pBeta, ws);
    k_final      <<<NSEG,    32, 0, stream>>>(r, pOmega, pAlpha, pBeta, ws, out);
}